// CenterLoss_3796751090214
// MI455X (gfx1250) — compile-verified
//
#include <hip/hip_runtime.h>

typedef __attribute__((ext_vector_type(2))) float v2f;
typedef __attribute__((ext_vector_type(8))) float v8f;

#define EMBED 512
#define ROWS_PER_BLOCK 16
// 512 + 4 pad dwords: row stride advances banks by 4 per row -> the strided
// per-lane ds_load_b64 pattern in phase 2 is bank-conflict-free.
#define LDS_STRIDE 516

__global__ void CenterLoss_init_kernel(float* out, float addConst) {
    if (threadIdx.x == 0 && blockIdx.x == 0) out[0] = addConst;
}

__global__ __launch_bounds__(32)
void CenterLoss_3796751090214_kernel(const float* __restrict__ x,
                                     const int* __restrict__ labels,
                                     const float* __restrict__ centers,
                                     float* __restrict__ out,
                                     float invB) {
    __shared__ float yds[ROWS_PER_BLOCK * LDS_STRIDE];
    const int lane = threadIdx.x;                 // wave32: 0..31
    const int rowBase = blockIdx.x * ROWS_PER_BLOCK;

    // ---- Phase 1: gather center rows, compute Y = x - c, stage into LDS.
    // Coalesced: 32 lanes x float4 = 512B contiguous per pass.
    for (int m = 0; m < ROWS_PER_BLOCK; ++m) {
        const int r = rowBase + m;
        const int lbl = labels[r];
        const float* xr = x + (size_t)r * EMBED;
        const float* cr = centers + (size_t)lbl * EMBED;
        float* yr = &yds[m * LDS_STRIDE];
#pragma unroll
        for (int p = 0; p < EMBED / 128; ++p) {
            const int c0 = p * 128 + lane * 4;
            const float4 xv = *(const float4*)(xr + c0);
            const float4 cv = *(const float4*)(cr + c0);
            float4 d;
            d.x = xv.x - cv.x;
            d.y = xv.y - cv.y;
            d.z = xv.z - cv.z;
            d.w = xv.w - cv.w;
            *(float4*)(yr + c0) = d;
        }
    }
    __syncthreads();

    // ---- Phase 2: D = Y * Y^T via V_WMMA_F32_16X16X4_F32; diag(D) = ||y_m||^2.
    // A-layout (16x4 f32): lane L holds row M = L&15; VGPR0/1 = K {0,1} for
    // lanes 0-15 and K {2,3} for lanes 16-31. B mirrors with lane = N, so
    // feeding identical registers to A and B computes Y * Y^T.
    const int m = lane & 15;
    const int khalf = (lane >> 4) << 1;           // 0 or 2
    const float* yrow = &yds[m * LDS_STRIDE + khalf];

    v8f acc0 = {};
    v8f acc1 = {};
#pragma unroll 4
    for (int k0 = 0; k0 < EMBED; k0 += 8) {
        const v2f a0 = *(const v2f*)(yrow + k0);
        const v2f a1 = *(const v2f*)(yrow + k0 + 4);
        acc0 = __builtin_amdgcn_wmma_f32_16x16x4_f32(
            false, a0, false, a0, (short)0, acc0, false, false);
        acc1 = __builtin_amdgcn_wmma_f32_16x16x4_f32(
            false, a1, false, a1, (short)0, acc1, false, false);
    }
    const v8f acc = acc0 + acc1;

    // ---- Phase 3: extract diagonal.
    // D[M][N]: VGPR g, lanes 0-15 -> M=g,  N=lane;
    //                  lanes 16-31 -> M=g+8, N=lane-16.
    // Diagonal: row g at lane g (g<8), row g+8 at lane g+24.
    const bool valid = (lane < 8) || (lane >= 24);
    const int idx = (lane < 8) ? lane : (lane - 24);
    float d = acc[0];
#pragma unroll
    for (int g = 1; g < 8; ++g) {
        if (idx == g) d = acc[g];
    }

    // Reference: clamp(dist, 1e-12, 1e12) on the selected entry.
    float t = valid ? fminf(fmaxf(d, 1e-12f), 1e12f) : 0.0f;

    // Wave32 reduction.
#pragma unroll
    for (int off = 16; off > 0; off >>= 1) {
        t += __shfl_xor(t, off, 32);
    }
    if (lane == 0) {
        atomicAdd(out, t * invB);
    }
}

extern "C" void kernel_launch(void* const* d_in, const int* in_sizes, int n_in,
                              void* d_out, int out_size, void* d_ws, size_t ws_size,
                              hipStream_t stream) {
    (void)n_in; (void)out_size; (void)d_ws; (void)ws_size;
    const float* x       = (const float*)d_in[0];
    const int*   labels  = (const int*)d_in[1];
    const float* centers = (const float*)d_in[2];
    float* out = (float*)d_out;

    const int B = in_sizes[1];                    // 4096
    const int embed = in_sizes[0] / B;            // 512
    const int C = in_sizes[2] / embed;            // 10000

    // Masked-out zeros clamp up to 1e-12: constant term (C-1)*1e-12.
    const float addConst = (float)((double)(C - 1) * 1e-12);
    const float invB = 1.0f / (float)B;

    CenterLoss_init_kernel<<<1, 32, 0, stream>>>(out, addConst);
    CenterLoss_3796751090214_kernel<<<B / ROWS_PER_BLOCK, 32, 0, stream>>>(
        x, labels, centers, out, invB);
}